// QuantumLayer_12635793785140
// MI455X (gfx1250) — compile-verified
//
#include <hip/hip_runtime.h>
#include <math.h>

// Problem constants
#define DCUT   8
#define NMODE  4
#define NLAY   2
#define BATCH  4096
#define NSTATE 4096              // 8^4
#define WSZ    16384             // 128x128 floats per beamsplitter W matrix
#define NBS    24                // 2 layers * 2 interferometers * 6 BS gates
#define SQ_OFF   (NBS * WSZ)     // squeeze gates: 8 * 128 floats
#define DISP_OFF (SQ_OFF + 1024) // displacement gates: 8 * 128 floats

typedef float v2f __attribute__((ext_vector_type(2)));
typedef float v8f __attribute__((ext_vector_type(8)));

// --------------------------------------------------------------------------
// Gate builder kernel.
//  blocks 0..23 : beamsplitter expm(64x64 complex) -> real 128x128 W matrix
//  blocks 24..39: squeeze / displacement expm(8x8 complex)
__global__ __launch_bounds__(256) void build_gates(
    const float* __restrict__ theta1, const float* __restrict__ phi1,
    const float* __restrict__ theta2, const float* __restrict__ phi2,
    const float* __restrict__ r_,     const float* __restrict__ phi_r,
    const float* __restrict__ a_amp,  const float* __restrict__ phi_a,
    float* __restrict__ wsF)
{
    extern __shared__ float sm[];   // builder uses up to 3 * 8192 floats (96 KB)
    const int tid = threadIdx.x;

    if (blockIdx.x < NBS) {
        // ---- Beamsplitter: H = theta*(e^{i phi} A (x) A^+  -  e^{-i phi} A^+ (x) A)
        const int bs  = blockIdx.x;
        const int l   = bs / 12;
        const int itf = (bs / 6) & 1;
        const int g   = bs % 6;
        const float theta = (itf ? theta2 : theta1)[l * 6 + g];
        const float phi   = (itf ? phi2   : phi1  )[l * 6 + g];
        const float cph = cosf(phi), sph = sinf(phi);

        float* X  = sm;                 // X = H / 2^8   (re at [e], im at [4096+e])
        float* A  = sm + 8192;          // accumulator
        float* T  = sm + 16384;         // temp
        const float inv = 1.0f / 256.0f;

        for (int e = tid; e < 4096; e += 256) {
            int row = e >> 6, col = e & 63;
            int i = row >> 3, j = row & 7, k = col >> 3, ll = col & 7;
            float hr = 0.f, hi = 0.f;
            if (k == i + 1 && ll == j - 1) {            // + e^{i phi} sqrt(k)sqrt(j)
                float c = theta * sqrtf((float)(k * j));
                hr += c * cph; hi += c * sph;
            }
            if (k == i - 1 && ll == j + 1) {            // - e^{-i phi} sqrt(i)sqrt(l)
                float c = theta * sqrtf((float)(i * ll));
                hr -= c * cph; hi += c * sph;
            }
            X[e] = hr * inv; X[4096 + e] = hi * inv;
            A[e] = (row == col) ? 1.f : 0.f; A[4096 + e] = 0.f;
        }
        __syncthreads();

        // Horner Taylor: A = I + X(I + X/2(I + ... ))
        for (int t = 12; t >= 1; --t) {
            for (int e = tid; e < 4096; e += 256) {
                int row = e >> 6, col = e & 63;
                float sr = 0.f, si = 0.f;
                for (int p = 0; p < 64; ++p) {
                    float xr = X[row * 64 + p],        xi = X[4096 + row * 64 + p];
                    float ar = A[p * 64 + col],        ai = A[4096 + p * 64 + col];
                    sr += xr * ar - xi * ai;
                    si += xr * ai + xi * ar;
                }
                T[e] = sr; T[4096 + e] = si;
            }
            __syncthreads();
            const float invt = 1.0f / (float)t;
            for (int e = tid; e < 4096; e += 256) {
                int row = e >> 6, col = e & 63;
                A[e]        = ((row == col) ? 1.f : 0.f) + T[e] * invt;
                A[4096 + e] = T[4096 + e] * invt;
            }
            __syncthreads();
        }

        // 8 squarings (ping-pong A <-> T)
        float* Ac = A; float* Tc = T;
        for (int q = 0; q < 8; ++q) {
            for (int e = tid; e < 4096; e += 256) {
                int row = e >> 6, col = e & 63;
                float sr = 0.f, si = 0.f;
                for (int p = 0; p < 64; ++p) {
                    float xr = Ac[row * 64 + p], xi = Ac[4096 + row * 64 + p];
                    float ar = Ac[p * 64 + col], ai = Ac[4096 + p * 64 + col];
                    sr += xr * ar - xi * ai;
                    si += xr * ai + xi * ar;
                }
                Tc[e] = sr; Tc[4096 + e] = si;
            }
            __syncthreads();
            float* t2 = Ac; Ac = Tc; Tc = t2;
        }

        // Expand G (64x64 complex) to W (128x128 real):
        //   W[o][k],  Out[r,o] = sum_k S[r,k] * W[o][k]
        float* Wout = wsF + bs * WSZ;
        for (int e = tid; e < WSZ; e += 256) {
            int o = e >> 7, kk = e & 127;
            int o2 = o & 63, k2 = kk & 63;
            float gr = Ac[o2 * 64 + k2], gi = Ac[4096 + o2 * 64 + k2];
            float val;
            if (o < 64) val = (kk < 64) ? gr : -gi;
            else        val = (kk < 64) ? gi :  gr;
            Wout[e] = val;
        }
    } else {
        // ---- Single-mode 8x8 expm (type 0 = squeeze, 1 = displacement)
        const int gi   = blockIdx.x - NBS;
        const int type = gi >> 3;
        const int g    = gi & 7;
        const int l    = g >> 2, m = g & 3;

        float* Xr = sm;       float* Xi = sm + 64;
        float* Ar = sm + 128; float* Ai = sm + 192;
        float* Tr = sm + 256; float* Ti = sm + 320;

        if (tid < 64) {
            int row = tid >> 3, col = tid & 7;
            float kr = 0.f, ki = 0.f;
            if (type == 0) {    // S(z) generator: 0.5*(conj(z) A^2 - z A+^2)
                float rr = r_[l * 4 + m], ph = phi_r[l * 4 + m];
                float zr = rr * cosf(ph), zi = rr * sinf(ph);
                if (col == row + 2) {
                    float c = 0.5f * sqrtf((float)((row + 1) * (row + 2)));
                    kr += c * zr; ki -= c * zi;
                }
                if (row == col + 2) {
                    float c = 0.5f * sqrtf((float)((col + 1) * (col + 2)));
                    kr -= c * zr; ki -= c * zi;
                }
            } else {            // D(a) generator: a A+ - conj(a) A
                float aa = a_amp[l * 4 + m], ph = phi_a[l * 4 + m];
                float ar = aa * cosf(ph), ai = aa * sinf(ph);
                if (row == col + 1) { float c = sqrtf((float)row); kr += c * ar; ki += c * ai; }
                if (col == row + 1) { float c = sqrtf((float)col); kr -= c * ar; ki += c * ai; }
            }
            Xr[tid] = kr; Xi[tid] = ki;
            Ar[tid] = (row == col) ? 1.f : 0.f; Ai[tid] = 0.f;
        }
        __syncthreads();
        for (int t = 16; t >= 1; --t) {
            if (tid < 64) {
                int row = tid >> 3, col = tid & 7;
                float sr = 0.f, si = 0.f;
                for (int p = 0; p < 8; ++p) {
                    float xr = Xr[row * 8 + p], xi = Xi[row * 8 + p];
                    float ar = Ar[p * 8 + col], ai = Ai[p * 8 + col];
                    sr += xr * ar - xi * ai;
                    si += xr * ai + xi * ar;
                }
                Tr[tid] = sr; Ti[tid] = si;
            }
            __syncthreads();
            if (tid < 64) {
                int row = tid >> 3, col = tid & 7;
                float invt = 1.0f / (float)t;
                Ar[tid] = ((row == col) ? 1.f : 0.f) + Tr[tid] * invt;
                Ai[tid] = Ti[tid] * invt;
            }
            __syncthreads();
        }
        if (tid < 64) {
            float* dst = wsF + (type ? DISP_OFF : SQ_OFF) + g * 128;
            dst[tid] = Ar[tid]; dst[64 + tid] = Ai[tid];
        }
    }
}

// --------------------------------------------------------------------------
// Beamsplitter apply via fp32 WMMA: Out[64x128] = S[64x128] * W^T
// PAIR is compile-time: all LDS indexing is constant-shift arithmetic.
//  PAIR 0 -> modes (0,1): idx = kp*64 + r          (KS = 64)
//  PAIR 1 -> modes (1,2): idx = base(r) + kp*8     (KS = 8)
//  PAIR 2 -> modes (2,3): idx = r*64 + kp          (KS = 1)
// 8 waves x (1 N-tile o=wave*16.., 4 M-tiles), K = 128 in steps of 4,
// split into re-plane (k<64) and im-plane (k>=64) segments.
template<int PAIR>
__device__ __attribute__((noinline)) void applyBS(float* sm, int cur,
                                                  const float* __restrict__ W)
{
    const int tid  = threadIdx.x;
    const int wave = tid >> 5;
    const int lane = tid & 31;
    const int lm   = lane & 15;
    const int half = lane >> 4;

    float* curRe = sm + cur * 8192;        float* curIm = curRe + NSTATE;
    float* outRe = sm + (cur ^ 1) * 8192;  float* outIm = outRe + NSTATE;
    __syncthreads();

    constexpr int KS = (PAIR == 0) ? 64 : ((PAIR == 1) ? 8 : 1);
    const int o = wave * 16 + lm;
    const float* Wrow = W + o * 128;

    int rbase[4];
#pragma unroll
    for (int mt = 0; mt < 4; ++mt) {
        const int r = mt * 16 + lm;
        rbase[mt] = (PAIR == 0) ? r
                  : (PAIR == 1) ? (((r & 0x38) << 6) | (r & 7))
                  : (r << 6);
    }

    v8f acc[4];
#pragma unroll
    for (int i = 0; i < 4; ++i) acc[i] = (v8f){0, 0, 0, 0, 0, 0, 0, 0};

#pragma unroll
    for (int hseg = 0; hseg < 2; ++hseg) {
        const float* plane = hseg ? curIm : curRe;
#pragma unroll
        for (int k0 = 0; k0 < 64; k0 += 4) {
            const int kb = k0 + half * 2;                        // segment-local K
            v2f Bv = *(const v2f*)(Wrow + hseg * 64 + kb);       // W^T 4xK x 16xN
#pragma unroll
            for (int mt = 0; mt < 4; ++mt) {
                v2f Av;
                if (KS == 1) {
                    Av = *(const v2f*)(plane + rbase[mt] + kb);  // ds_load_b64
                } else {
                    Av.x = plane[rbase[mt] + kb * KS];
                    Av.y = plane[rbase[mt] + kb * KS + KS];
                }
                acc[mt] = __builtin_amdgcn_wmma_f32_16x16x4_f32(
                    false, Av, false, Bv, (short)0, acc[mt], false, false);
            }
        }
    }

    // D layout: VGPR p, lanes0-15 -> M=p, lanes16-31 -> M=8+p, N=lane&15
    float* oplane = (o < 64) ? outRe : outIm;                    // uniform per wave
    const int oc = o & 63;
#pragma unroll
    for (int mt = 0; mt < 4; ++mt)
#pragma unroll
        for (int p = 0; p < 8; ++p) {
            const int row = mt * 16 + half * 8 + p;
            const int idx = (PAIR == 0) ? ((oc << 6) | row)
                          : (PAIR == 1) ? (((row & 0x38) << 6) | (oc << 3) | (row & 7))
                          : ((row << 6) | oc);
            oplane[idx] = acc[mt][p];
        }
    __syncthreads();
}

// Single-mode 8x8 complex gate (VALU path).
__device__ __attribute__((noinline)) void applyG1(float* sm, int cur,
                                                  const float* __restrict__ g, int m,
                                                  float* gBuf)
{
    const int tid = threadIdx.x;
    float* curRe = sm + cur * 8192;        float* curIm = curRe + NSTATE;
    float* outRe = sm + (cur ^ 1) * 8192;  float* outIm = outRe + NSTATE;
    __syncthreads();
    if (tid < 128) gBuf[tid] = g[tid];
    __syncthreads();

    const int s = 1 << ((3 - m) * 3);
    int os[3]; { int j = 0; for (int mm = 0; mm < 4; ++mm) if (mm != m) os[j++] = 1 << ((3 - mm) * 3); }

    for (int rid = tid; rid < 512; rid += 256) {
        int base = (rid >> 6) * os[0] + ((rid >> 3) & 7) * os[1] + (rid & 7) * os[2];
        float sr[8], si[8];
#pragma unroll
        for (int k = 0; k < 8; ++k) { int id = base + k * s; sr[k] = curRe[id]; si[k] = curIm[id]; }
#pragma unroll
        for (int oo = 0; oo < 8; ++oo) {
            float ar = 0.f, ai = 0.f;
#pragma unroll
            for (int k = 0; k < 8; ++k) {
                float gr = gBuf[oo * 8 + k], gi = gBuf[64 + oo * 8 + k];
                ar += gr * sr[k] - gi * si[k];
                ai += gr * si[k] + gi * sr[k];
            }
            int id = base + oo * s;
            outRe[id] = ar; outIm[id] = ai;
        }
    }
    __syncthreads();
}

// Fused diagonal phases: theta(idx) = sum_m v[m]*f(n_m), f(n)=n or n^2 (Kerr).
__device__ __attribute__((noinline)) void phasePass(float* sm, int cur,
                                                    const float* __restrict__ v, bool kerr)
{
    const int tid = threadIdx.x;
    float* curRe = sm + cur * 8192; float* curIm = curRe + NSTATE;
    const float v0 = v[0], v1 = v[1], v2 = v[2], v3 = v[3];
    __syncthreads();
    for (int idx = tid; idx < NSTATE; idx += 256) {
        float n0 = (float)((idx >> 9) & 7), n1 = (float)((idx >> 6) & 7);
        float n2 = (float)((idx >> 3) & 7), n3 = (float)(idx & 7);
        float th = kerr ? (v0*n0*n0 + v1*n1*n1 + v2*n2*n2 + v3*n3*n3)
                        : (v0*n0 + v1*n1 + v2*n2 + v3*n3);
        float cs = cosf(th), sn = sinf(th);
        float re = curRe[idx], im = curIm[idx];
        curRe[idx] = re * cs - im * sn;
        curIm[idx] = re * sn + im * cs;
    }
    __syncthreads();
}

// --------------------------------------------------------------------------
// Main evolution: one block per batch element, state resident in LDS.
__global__ __launch_bounds__(256) void evolve(
    const float* __restrict__ x,
    const float* __restrict__ varphi1, const float* __restrict__ varphi2,
    const float* __restrict__ kappa,
    const float* __restrict__ wsF, float* __restrict__ out)
{
    extern __shared__ float sm[];          // 2 * (re,im) planes = 64 KB
    __shared__ float colB[4][8];
    __shared__ float gBuf[128];
    __shared__ float red[4];

    const int b = blockIdx.x, tid = threadIdx.x;

    // Displacement embedding column: expm(alpha*(A+ - A)) e0, bidiagonal Taylor
    if (tid < 4) {
        float alpha = x[b * 4 + tid];
        float v[8], col[8];
#pragma unroll
        for (int n = 0; n < 8; ++n) { v[n] = (n == 0) ? 1.f : 0.f; col[n] = v[n]; }
        for (int t = 1; t <= 24; ++t) {
            float nv[8];
#pragma unroll
            for (int n = 0; n < 8; ++n) {
                float up = (n > 0) ? sqrtf((float)n) * v[n - 1] : 0.f;
                float dn = (n < 7) ? sqrtf((float)(n + 1)) * v[n + 1] : 0.f;
                nv[n] = alpha * (up - dn) / (float)t;
            }
#pragma unroll
            for (int n = 0; n < 8; ++n) { v[n] = nv[n]; col[n] += v[n]; }
        }
#pragma unroll
        for (int n = 0; n < 8; ++n) colB[tid][n] = col[n];
        red[tid] = 0.f;
    }
    __syncthreads();

    for (int idx = tid; idx < NSTATE; idx += 256) {
        sm[idx] = colB[0][idx >> 9] * colB[1][(idx >> 6) & 7]
                * colB[2][(idx >> 3) & 7] * colB[3][idx & 7];
        sm[NSTATE + idx] = 0.f;
    }

    int cur = 0;
#pragma unroll 1
    for (int l = 0; l < NLAY; ++l) {
        // interferometer 1 (mesh order: pairs 0,2,1,0,2,1)
        const float* W1 = wsF + (l * 2 + 0) * 6 * WSZ;
        applyBS<0>(sm, cur, W1 + 0 * WSZ); cur ^= 1;
        applyBS<2>(sm, cur, W1 + 1 * WSZ); cur ^= 1;
        applyBS<1>(sm, cur, W1 + 2 * WSZ); cur ^= 1;
        applyBS<0>(sm, cur, W1 + 3 * WSZ); cur ^= 1;
        applyBS<2>(sm, cur, W1 + 4 * WSZ); cur ^= 1;
        applyBS<1>(sm, cur, W1 + 5 * WSZ); cur ^= 1;
        phasePass(sm, cur, varphi1 + l * 4, false);
        for (int m = 0; m < 4; ++m) { applyG1(sm, cur, wsF + SQ_OFF + (l*4+m)*128, m, gBuf); cur ^= 1; }
        // interferometer 2
        const float* W2 = wsF + (l * 2 + 1) * 6 * WSZ;
        applyBS<0>(sm, cur, W2 + 0 * WSZ); cur ^= 1;
        applyBS<2>(sm, cur, W2 + 1 * WSZ); cur ^= 1;
        applyBS<1>(sm, cur, W2 + 2 * WSZ); cur ^= 1;
        applyBS<0>(sm, cur, W2 + 3 * WSZ); cur ^= 1;
        applyBS<2>(sm, cur, W2 + 4 * WSZ); cur ^= 1;
        applyBS<1>(sm, cur, W2 + 5 * WSZ); cur ^= 1;
        phasePass(sm, cur, varphi2 + l * 4, false);
        for (int m = 0; m < 4; ++m) { applyG1(sm, cur, wsF + DISP_OFF + (l*4+m)*128, m, gBuf); cur ^= 1; }
        phasePass(sm, cur, kappa + l * 4, true);
    }

    // <X_i> = 2 Re <psi| a_i |psi>
    float* curRe = sm + cur * 8192; float* curIm = curRe + NSTATE;
    float p[4] = {0.f, 0.f, 0.f, 0.f};
    for (int idx = tid; idx < NSTATE; idx += 256) {
        float re = curRe[idx], im = curIm[idx];
#pragma unroll
        for (int i = 0; i < 4; ++i) {
            int sh = (3 - i) * 3;
            int n = (idx >> sh) & 7;
            int idx2 = (idx + (1 << sh)) & (NSTATE - 1);
            float w = (n < 7) ? sqrtf((float)(n + 1)) : 0.f;
            p[i] += w * (re * curRe[idx2] + im * curIm[idx2]);
        }
    }
#pragma unroll
    for (int i = 0; i < 4; ++i) atomicAdd(&red[i], p[i]);
    __syncthreads();
    if (tid < 4) out[b * 4 + tid] = 2.f * red[tid];
}

// --------------------------------------------------------------------------
extern "C" void kernel_launch(void* const* d_in, const int* in_sizes, int n_in,
                              void* d_out, int out_size, void* d_ws, size_t ws_size,
                              hipStream_t stream)
{
    const float* x       = (const float*)d_in[0];
    const float* theta1  = (const float*)d_in[1];
    const float* phi1    = (const float*)d_in[2];
    const float* varphi1 = (const float*)d_in[3];
    const float* r_      = (const float*)d_in[4];
    const float* phi_r   = (const float*)d_in[5];
    const float* theta2  = (const float*)d_in[6];
    const float* phi2    = (const float*)d_in[7];
    const float* varphi2 = (const float*)d_in[8];
    const float* a_amp   = (const float*)d_in[9];
    const float* phi_a   = (const float*)d_in[10];
    const float* kappa   = (const float*)d_in[11];
    float* out = (float*)d_out;
    float* wsF = (float*)d_ws;

    (void)in_sizes; (void)n_in; (void)out_size; (void)ws_size;

    hipFuncSetAttribute((const void*)build_gates,
                        hipFuncAttributeMaxDynamicSharedMemorySize, 96 * 1024);
    hipFuncSetAttribute((const void*)evolve,
                        hipFuncAttributeMaxDynamicSharedMemorySize, 64 * 1024);

    build_gates<<<40, 256, 96 * 1024, stream>>>(theta1, phi1, theta2, phi2,
                                                r_, phi_r, a_amp, phi_a, wsF);
    evolve<<<BATCH, 256, 64 * 1024, stream>>>(x, varphi1, varphi2, kappa, wsF, out);
}